// v_ParallelTransportPool_22746146799942
// MI455X (gfx1250) — compile-verified
//
#include <hip/hip_runtime.h>

#define NSITE (32 * 16 * 16 * 16) /* 131072 sites: T=32, X=Y=Z=16 */

typedef __attribute__((ext_vector_type(2))) float v2f;
typedef __attribute__((ext_vector_type(8))) float v8f;

// ---------------------------------------------------------------------------
// Kernel 1: per-site complex spin transform g = w * f via V_WMMA_F32_16X16X4_F32
//
// Realified: W2 = [[wr, -wi],[wi, wr]] (8x8), F2 = [fr; fi] (8x3), W2*F2 = [gr; gi].
// Each wave handles 2 sites (sA rows 0-7, sB rows 8-15 of the A matrix).
// K=8 split into two 16x16x4 WMMAs. B cols 0-2 = site sA, cols 3-5 = site sB.
// ---------------------------------------------------------------------------
__global__ __launch_bounds__(256) void spin_transform_wmma(
    const float* __restrict__ f_re, const float* __restrict__ f_im,
    const float* __restrict__ w_re, const float* __restrict__ w_im,
    float* __restrict__ g_re, float* __restrict__ g_im)
{
    const int lane = threadIdx.x & 31;
    const int wave = blockIdx.x * (blockDim.x >> 5) + (threadIdx.x >> 5);
    const int sA = wave * 2;
    const int sB = sA + 1;

    // ---- A operand (16x4 fp32): lanes 0-15 -> M=lane, K={0,1};
    //                             lanes 16-31 -> M=lane-16, K={2,3}
    const int M  = lane & 15;
    const int k0 = (lane >> 4) << 1;       // 0 or 2
    const int aSite = (M < 8) ? sA : sB;
    const int r = M & 7;                   // row within W2
    const int i = r & 3;                   // spin row
    const bool rowIm = (r >= 4);

    const float* wr = w_re + (size_t)aSite * 16;
    const float* wi = w_im + (size_t)aSite * 16;
    const float wr0 = wr[i * 4 + k0], wr1 = wr[i * 4 + k0 + 1];
    const float wi0 = wi[i * 4 + k0], wi1 = wi[i * 4 + k0 + 1];

    v2f a0, a1;
    a0.x = rowIm ? wi0 : wr0;              // W2 columns 0..3
    a0.y = rowIm ? wi1 : wr1;
    a1.x = rowIm ? wr0 : -wi0;             // W2 columns 4..7
    a1.y = rowIm ? wr1 : -wi1;

    // ---- B operand (4x16 fp32): rows = F2 rows (K), columns = packed sites
    const int N = lane & 15;
    v2f b0 = {0.f, 0.f};                   // F2 rows 0-3 (= f_re)
    v2f b1 = {0.f, 0.f};                   // F2 rows 4-7 (= f_im)
    if (N < 6) {
        const int bSite = (N < 3) ? sA : sB;
        const int c     = (N < 3) ? N : (N - 3);
        const float* fr = f_re + (size_t)bSite * 12;
        const float* fi = f_im + (size_t)bSite * 12;
        b0.x = fr[k0 * 3 + c];  b0.y = fr[(k0 + 1) * 3 + c];
        b1.x = fi[k0 * 3 + c];  b1.y = fi[(k0 + 1) * 3 + c];
    }

    // D = A0*B0 + A1*B1  (full-EXEC, top-level — no divergence here)
    v8f d = {};
    d = __builtin_amdgcn_wmma_f32_16x16x4_f32(false, a0, false, b0, (short)0, d, false, false);
    d = __builtin_amdgcn_wmma_f32_16x16x4_f32(false, a1, false, b1, (short)0, d, false, false);

    // ---- Store block-diagonal results. C/D layout: VGPR r holds row M=r
    // (lanes 0-15) and M=r+8 (lanes 16-31), N = lane%16.
    const bool hi    = (lane >= 16);
    const int  oSite = hi ? sB : sA;
    const int  cc    = hi ? (N - 3) : N;
    const bool active = hi ? (N >= 3 && N < 6) : (N < 3);
    if (active) {
        float* gr = g_re + (size_t)oSite * 12;
        float* gi = g_im + (size_t)oSite * 12;
        gr[0 + cc] = d[0]; gr[3 + cc] = d[1]; gr[6 + cc] = d[2]; gr[9 + cc] = d[3];
        gi[0 + cc] = d[4]; gi[3 + cc] = d[5]; gi[6 + cc] = d[6]; gi[9 + cc] = d[7];
    }
}

// ---------------------------------------------------------------------------
// Complex 3x3 color matmul accumulate: acc += U * g  (or U^H * g if DAG).
// ---------------------------------------------------------------------------
template <bool DAG>
__device__ __forceinline__ void color_hop(const float* __restrict__ ur,
                                          const float* __restrict__ ui,
                                          const float* __restrict__ nr,
                                          const float* __restrict__ ni,
                                          float* accR, float* accI)
{
#pragma unroll
    for (int s = 0; s < 4; ++s)
#pragma unroll
        for (int a = 0; a < 3; ++a) {
            float cr = 0.f, ci = 0.f;
#pragma unroll
            for (int b = 0; b < 3; ++b) {
                const int ue = DAG ? (b * 3 + a) : (a * 3 + b);
                const float Ur = ur[ue];
                const float Ui = DAG ? -ui[ue] : ui[ue];   // conj for dagger
                const float Gr = nr[s * 3 + b], Gi = ni[s * 3 + b];
                cr = __builtin_fmaf(Ur, Gr, cr);
                cr = __builtin_fmaf(-Ui, Gi, cr);
                ci = __builtin_fmaf(Ur, Gi, ci);
                ci = __builtin_fmaf(Ui, Gr, ci);
            }
            accR[s * 3 + a] += cr;
            accI[s * 3 + a] += ci;
        }
}

// ---------------------------------------------------------------------------
// Kernel 2: out(x) = g(x) + sum_mu U_mu(x) g(x+mu) + sum_mu U_mu(x-mu)^H g(x-mu)
//
// One block = one (t,x) plane of 16x16 (y,z) sites. Since y,z are the low 8
// site bits, ALL y/z-direction neighbors (with periodic wrap) live inside the
// block's tile -> stage the tile's g in LDS (24 KB of the 320 KB WGP LDS).
// Identity + 4 of 8 hops then read from LDS; t/x hops and U stay on the
// global path (L2-resident working set). Output interleaved complex64.
// ---------------------------------------------------------------------------
__global__ __launch_bounds__(256) void hop_accumulate(
    const float* __restrict__ g_re, const float* __restrict__ g_im,
    const float* __restrict__ U_re, const float* __restrict__ U_im,
    float* __restrict__ out)
{
    extern __shared__ float smem[];        // [2][256][12] floats = 24 KB
    float* sgr = smem;
    float* sgi = smem + 256 * 12;

    const int tid   = threadIdx.x;         // = (y<<4)|z within the tile
    const int tile0 = blockIdx.x << 8;     // first site of this (t,x) tile
    const int site  = tile0 + tid;

    // ---- cooperative, fully-coalesced tile load of g into LDS ----
    {
        const float4* gr4 = (const float4*)(g_re + (size_t)tile0 * 12);
        const float4* gi4 = (const float4*)(g_im + (size_t)tile0 * 12);
        float4* sr4 = (float4*)sgr;
        float4* si4 = (float4*)sgi;
#pragma unroll
        for (int k = 0; k < 3; ++k) {      // 768 float4 per array / 256 threads
            sr4[tid + 256 * k] = gr4[tid + 256 * k];
            si4[tid + 256 * k] = gi4[tid + 256 * k];
        }
    }
    __syncthreads();

    const int z = site & 15;
    const int y = (site >> 4) & 15;
    const int x = (site >> 8) & 15;
    const int t = site >> 12;

    const int tp = (t + 1) & 31,  tm = (t + 31) & 31;
    const int xp = (x + 1) & 15,  xm = (x + 15) & 15;
    const int yp = (y + 1) & 15,  ym = (y + 15) & 15;
    const int zp = (z + 1) & 15,  zm = (z + 15) & 15;

    // global neighbor sites for t,x hops
    const int spT = (tp << 12) | (site & 0x0FFF);
    const int smT = (tm << 12) | (site & 0x0FFF);
    const int spX = (site & ~0xF00) | (xp << 8);
    const int smX = (site & ~0xF00) | (xm << 8);
    // in-tile local indices for y,z hops
    const int lyp = (yp << 4) | (tid & 0x0F);
    const int lym = (ym << 4) | (tid & 0x0F);
    const int lzp = (tid & 0xF0) | zp;
    const int lzm = (tid & 0xF0) | zm;

    float accR[12], accI[12];
#pragma unroll
    for (int k = 0; k < 12; ++k) {         // identity term from LDS
        accR[k] = sgr[tid * 12 + k];
        accI[k] = sgi[tid * 12 + k];
    }

    // ---- mu = 0 (t) and mu = 1 (x): neighbors from global ----
    {
        const size_t u0 = ((size_t)0 * NSITE + site) * 9;
        color_hop<false>(U_re + u0, U_im + u0,
                         g_re + (size_t)spT * 12, g_im + (size_t)spT * 12, accR, accI);
        const size_t u0m = ((size_t)0 * NSITE + smT) * 9;
        color_hop<true>(U_re + u0m, U_im + u0m,
                        g_re + (size_t)smT * 12, g_im + (size_t)smT * 12, accR, accI);

        const size_t u1 = ((size_t)1 * NSITE + site) * 9;
        color_hop<false>(U_re + u1, U_im + u1,
                         g_re + (size_t)spX * 12, g_im + (size_t)spX * 12, accR, accI);
        const size_t u1m = ((size_t)1 * NSITE + smX) * 9;
        color_hop<true>(U_re + u1m, U_im + u1m,
                        g_re + (size_t)smX * 12, g_im + (size_t)smX * 12, accR, accI);
    }

    // ---- mu = 2 (y) and mu = 3 (z): neighbors from LDS ----
    {
        const size_t u2 = ((size_t)2 * NSITE + site) * 9;
        color_hop<false>(U_re + u2, U_im + u2,
                         sgr + lyp * 12, sgi + lyp * 12, accR, accI);
        const size_t u2m = ((size_t)2 * NSITE + (tile0 + lym)) * 9;
        color_hop<true>(U_re + u2m, U_im + u2m,
                        sgr + lym * 12, sgi + lym * 12, accR, accI);

        const size_t u3 = ((size_t)3 * NSITE + site) * 9;
        color_hop<false>(U_re + u3, U_im + u3,
                         sgr + lzp * 12, sgi + lzp * 12, accR, accI);
        const size_t u3m = ((size_t)3 * NSITE + (tile0 + lzm)) * 9;
        color_hop<true>(U_re + u3m, U_im + u3m,
                        sgr + lzm * 12, sgi + lzm * 12, accR, accI);
    }

    // interleaved complex64 output: [site][spin][color]{re,im}
    float* o = out + (size_t)site * 24;
#pragma unroll
    for (int k = 0; k < 12; ++k) {
        o[2 * k]     = accR[k];
        o[2 * k + 1] = accI[k];
    }
}

extern "C" void kernel_launch(void* const* d_in, const int* in_sizes, int n_in,
                              void* d_out, int out_size, void* d_ws, size_t ws_size,
                              hipStream_t stream) {
    const float* f_re = (const float*)d_in[0];
    const float* f_im = (const float*)d_in[1];
    const float* w_re = (const float*)d_in[2];
    const float* w_im = (const float*)d_in[3];
    const float* U_re = (const float*)d_in[4];
    const float* U_im = (const float*)d_in[5];
    float* out  = (float*)d_out;

    float* g_re = (float*)d_ws;                       // NSITE*12 floats
    float* g_im = g_re + (size_t)NSITE * 12;          // NSITE*12 floats (~12.6 MB total)

    // 8 waves/block * 2 sites/wave = 16 sites/block -> 8192 blocks, every wave full.
    spin_transform_wmma<<<NSITE / 16, 256, 0, stream>>>(f_re, f_im, w_re, w_im, g_re, g_im);

    // one block per (t,x) plane; 24 KB dynamic LDS for the g tile
    hop_accumulate<<<NSITE / 256, 256, 2 * 256 * 12 * sizeof(float), stream>>>(
        g_re, g_im, U_re, U_im, out);
}